// GATGraphClassifier_1391569404378
// MI455X (gfx1250) — compile-verified
//
#include <hip/hip_runtime.h>
#include <math.h>
#include <stdint.h>

#define NN      50000           // nodes
#define NE      800000          // edges (without self loops)
#define EP      (NE + NN)       // edges + self loops
#define INCH    128
#define HIDC    64
#define NHEADS  4
#define F1      (NHEADS * HIDC) // 256
#define OUTC    10
#define NGRAPH  500
#define NEG_SLOPE 0.2f

typedef float v2f __attribute__((ext_vector_type(2)));
typedef float v8f __attribute__((ext_vector_type(8)));

// ---- monotonic float <-> uint key for atomic segment-max ----
__device__ __forceinline__ unsigned f32key(float f) {
  unsigned u = __float_as_uint(f);
  return u ^ ((unsigned)((int)u >> 31) | 0x80000000u);
}
__device__ __forceinline__ float keyf32(unsigned u) {
  unsigned b = (u & 0x80000000u) ? (u ^ 0x80000000u) : ~u;
  return __uint_as_float(b);
}

// ---------------------------------------------------------------------------
// f32 GEMM on the WMMA pipe: C[M,N] = A[M,K] @ B[K,N], all row-major.
// One wave -> one 16x16 C tile, chained V_WMMA_F32_16X16X4_F32 over K.
// Fragment layouts per CDNA5 ISA 7.12.2:
//   A 16x4 : lane l, vgpr v -> m = l%16, k = 2*(l/16)+v
//   B 4x16 : lane l, vgpr v -> n = l%16, k = 2*(l/16)+v
//   C 16x16: lane l, vgpr r -> n = l%16, m = r + 8*(l/16)
// M,K,N must be multiples of 16/4/16 (true for all calls here).
// ---------------------------------------------------------------------------
__global__ void gemm_wmma_f32(const float* __restrict__ A,
                              const float* __restrict__ B,
                              float* __restrict__ C,
                              int M, int K, int N)
{
  const int wave = (int)((blockIdx.x * blockDim.x + threadIdx.x) >> 5);
  const int ntn  = N >> 4;
  const int tm   = wave / ntn;
  const int tn   = wave % ntn;
  if (tm * 16 >= M) return;
  const int lane = threadIdx.x & 31;
  const int lh   = lane >> 4;   // 0/1: which half of the wave
  const int lm   = lane & 15;

  const float* Arow = A + (size_t)(tm * 16 + lm) * K;
  const float* Bcol = B + tn * 16 + lm;

  v8f acc = {};
  for (int k0 = 0; k0 < K; k0 += 4) {
    const int ka = k0 + 2 * lh;
    v2f a, b;
    a.x = Arow[ka];
    a.y = Arow[ka + 1];
    b.x = Bcol[(size_t)ka * N];
    b.y = Bcol[(size_t)(ka + 1) * N];
    acc = __builtin_amdgcn_wmma_f32_16x16x4_f32(
        false, a, false, b, (short)0, acc, false, false);
  }
  float* Cout = C + tn * 16 + lm;
#pragma unroll
  for (int r = 0; r < 8; ++r) {
    const int row = tm * 16 + r + 8 * lh;
    Cout[(size_t)row * N] = acc[r];
  }
}

// al_s[n,h] = <h[n,h,:], a_src[h,:]> ; al_d likewise
__global__ void attn_dots(const float* __restrict__ h,
                          const float* __restrict__ a_src,
                          const float* __restrict__ a_dst,
                          float* __restrict__ als, float* __restrict__ ald,
                          int N, int H, int C)
{
  int idx = blockIdx.x * blockDim.x + threadIdx.x;
  if (idx >= N * H) return;
  int n = idx / H, hh = idx % H;
  const float* hp = h + (size_t)n * H * C + hh * C;
  const float* as = a_src + hh * C;
  const float* ad = a_dst + hh * C;
  float s = 0.f, d = 0.f;
  for (int c = 0; c < C; ++c) { float v = hp[c]; s = fmaf(v, as[c], s); d = fmaf(v, ad[c], d); }
  als[idx] = s; ald[idx] = d;
}

// e = leaky_relu(al_s[src] + al_d[dst]); store e; atomic segment-max over dst
__global__ void edge_logits(const int* __restrict__ srcA, const int* __restrict__ dstA,
                            const float* __restrict__ als, const float* __restrict__ ald,
                            float* __restrict__ elog, unsigned* __restrict__ emax, int H)
{
  long idx = (long)blockIdx.x * blockDim.x + threadIdx.x;
  if (idx >= (long)EP * H) return;
  int e = (int)(idx / H), hh = (int)(idx % H);
  int s, d;
  if (e < NE) { s = srcA[e]; d = dstA[e]; } else { s = d = e - NE; }
  float v = als[s * H + hh] + ald[d * H + hh];
  v = v > 0.f ? v : NEG_SLOPE * v;
  elog[idx] = v;
  atomicMax(&emax[d * H + hh], f32key(v));
}

// ex = exp(e - max[dst]); store in place; atomic segment-sum denom
__global__ void edge_exp(const int* __restrict__ dstA,
                         float* __restrict__ elog, const unsigned* __restrict__ emax,
                         float* __restrict__ denom, int H)
{
  long idx = (long)blockIdx.x * blockDim.x + threadIdx.x;
  if (idx >= (long)EP * H) return;
  int e = (int)(idx / H), hh = (int)(idx % H);
  int d = (e < NE) ? dstA[e] : (e - NE);
  float ex = expf(elog[idx] - keyf32(emax[d * H + hh]));
  elog[idx] = ex;
  atomicAdd(&denom[d * H + hh], ex);
}

// out[dst, h*C+c] += h[src, h*C+c] * ex / (denom[dst,h] + 1e-16)
// F = H*C channels of one edge handled by F consecutive threads (coalesced).
__global__ void aggregate(const int* __restrict__ srcA, const int* __restrict__ dstA,
                          const float* __restrict__ h, const float* __restrict__ ex,
                          const float* __restrict__ denom,
                          float* __restrict__ out, int H, int C)
{
  const int F   = H * C;
  const int epb = blockDim.x / F;                     // edges per block
  long e = (long)blockIdx.x * epb + threadIdx.x / F;
  if (e >= EP) return;
  const int c  = threadIdx.x % F;
  const int hh = c / C;
  int s, d;
  if (e < NE) { s = srcA[e]; d = dstA[e]; } else { s = d = (int)e - NE; }
  float a = ex[e * H + hh] / (denom[d * H + hh] + 1e-16f);
  atomicAdd(&out[(size_t)d * F + c], h[(size_t)s * F + c] * a);
}

__global__ void bias_relu(float* __restrict__ x, const float* __restrict__ b, long n, int F)
{
  long i = (long)blockIdx.x * blockDim.x + threadIdx.x;
  if (i >= n) return;
  float v = x[i] + b[i % F];
  x[i] = v > 0.f ? v : 0.f;
}

// pooled[batch[n], c] += out2[n,c] + b2[c]
__global__ void pool_add(const float* __restrict__ out2, const float* __restrict__ b2,
                         const int* __restrict__ batch, float* __restrict__ pooled)
{
  long i = (long)blockIdx.x * blockDim.x + threadIdx.x;
  if (i >= (long)NN * HIDC) return;
  int n = (int)(i / HIDC), c = (int)(i % HIDC);
  atomicAdd(&pooled[(size_t)batch[n] * HIDC + c], out2[i] + b2[c]);
}

// logits = pooled @ fc_w + fc_b ; log_softmax. One thread per graph (tiny).
__global__ void head(const float* __restrict__ pooled, const float* __restrict__ fcw,
                     const float* __restrict__ fcb, float* __restrict__ out)
{
  int g = blockIdx.x * blockDim.x + threadIdx.x;
  if (g >= NGRAPH) return;
  float logit[OUTC];
  for (int o = 0; o < OUTC; ++o) {
    float s = fcb[o];
    for (int c = 0; c < HIDC; ++c) s = fmaf(pooled[g * HIDC + c], fcw[c * OUTC + o], s);
    logit[o] = s;
  }
  float m = logit[0];
  for (int o = 1; o < OUTC; ++o) m = fmaxf(m, logit[o]);
  float sum = 0.f;
  for (int o = 0; o < OUTC; ++o) sum += expf(logit[o] - m);
  float lse = m + logf(sum);
  for (int o = 0; o < OUTC; ++o) out[g * OUTC + o] = logit[o] - lse;
}

extern "C" void kernel_launch(void* const* d_in, const int* in_sizes, int n_in,
                              void* d_out, int out_size, void* d_ws, size_t ws_size,
                              hipStream_t stream) {
  (void)in_sizes; (void)n_in; (void)out_size; (void)ws_size;

  const float* x      = (const float*)d_in[0];
  const int*   edges  = (const int*)d_in[1];     // [2, NE] flat
  const int*   batch  = (const int*)d_in[2];
  const float* W1     = (const float*)d_in[3];
  const float* asrc1  = (const float*)d_in[4];
  const float* adst1  = (const float*)d_in[5];
  const float* b1     = (const float*)d_in[6];
  const float* W2     = (const float*)d_in[7];
  const float* asrc2  = (const float*)d_in[8];
  const float* adst2  = (const float*)d_in[9];
  const float* b2     = (const float*)d_in[10];
  const float* fcw    = (const float*)d_in[11];
  const float* fcb    = (const float*)d_in[12];
  float* out = (float*)d_out;

  const int* srcA = edges;
  const int* dstA = edges + NE;

  // ---- workspace layout (floats) ----
  float* ws = (float*)d_ws;
  size_t o = 0;
  float* h1    = ws + o; o += (size_t)NN * F1;     // x @ W1
  float* h2    = ws + o; o += (size_t)NN * HIDC;   // hrelu @ W2
  float* als1  = ws + o; o += (size_t)NN * NHEADS;
  float* ald1  = ws + o; o += (size_t)NN * NHEADS;
  float* e1    = ws + o; o += (size_t)EP * NHEADS; // logits -> ex
  float* als2  = ws + o; o += (size_t)NN;
  float* ald2  = ws + o; o += (size_t)NN;
  float* e2    = ws + o; o += (size_t)EP;
  // ---- zero-initialized accumulator region (contiguous, one memset) ----
  float* zbeg  = ws + o;
  float* out1  = ws + o; o += (size_t)NN * F1;
  float* out2  = ws + o; o += (size_t)NN * HIDC;
  unsigned* emax1 = (unsigned*)(ws + o); o += (size_t)NN * NHEADS;
  float* denom1   = ws + o; o += (size_t)NN * NHEADS;
  unsigned* emax2 = (unsigned*)(ws + o); o += (size_t)NN;
  float* denom2   = ws + o; o += (size_t)NN;
  float* pooled   = ws + o; o += (size_t)NGRAPH * HIDC;
  size_t zbytes = (size_t)((ws + o) - zbeg) * sizeof(float);

  hipMemsetAsync(zbeg, 0, zbytes, stream);   // 0u is also the key-min for emax

  const int T = 256;

  // ===== Layer 1 (H=4, C=64) =====
  {
    int waves = (NN / 16) * (F1 / 16);                     // 50000
    gemm_wmma_f32<<<(waves * 32 + T - 1) / T, T, 0, stream>>>(x, W1, h1, NN, INCH, F1);
  }
  attn_dots<<<(NN * NHEADS + T - 1) / T, T, 0, stream>>>(h1, asrc1, adst1, als1, ald1, NN, NHEADS, HIDC);
  {
    long n = (long)EP * NHEADS;
    edge_logits<<<(unsigned)((n + T - 1) / T), T, 0, stream>>>(srcA, dstA, als1, ald1, e1, emax1, NHEADS);
    edge_exp<<<(unsigned)((n + T - 1) / T), T, 0, stream>>>(dstA, e1, emax1, denom1, NHEADS);
  }
  aggregate<<<EP, T, 0, stream>>>(srcA, dstA, h1, e1, denom1, out1, NHEADS, HIDC); // 1 edge/block (F=256)
  bias_relu<<<(int)(((long)NN * F1 + T - 1) / T), T, 0, stream>>>(out1, b1, (long)NN * F1, F1);

  // ===== Layer 2 (H=1, C=64) =====
  {
    int waves = (NN / 16) * (HIDC / 16);                   // 12500
    gemm_wmma_f32<<<(waves * 32 + T - 1) / T, T, 0, stream>>>(out1, W2, h2, NN, F1, HIDC);
  }
  attn_dots<<<(NN + T - 1) / T, T, 0, stream>>>(h2, asrc2, adst2, als2, ald2, NN, 1, HIDC);
  edge_logits<<<(EP + T - 1) / T, T, 0, stream>>>(srcA, dstA, als2, ald2, e2, emax2, 1);
  edge_exp<<<(EP + T - 1) / T, T, 0, stream>>>(dstA, e2, emax2, denom2, 1);
  aggregate<<<(EP + 3) / 4, T, 0, stream>>>(srcA, dstA, h2, e2, denom2, out2, 1, HIDC); // 4 edges/block

  // ===== Pool + classifier head =====
  pool_add<<<(int)(((long)NN * HIDC + T - 1) / T), T, 0, stream>>>(out2, b2, batch, pooled);
  head<<<(NGRAPH + T - 1) / T, T, 0, stream>>>(pooled, fcw, fcb, out);
}